// GNN_26482768347970
// MI455X (gfx1250) — compile-verified
//
#include <hip/hip_runtime.h>
#include <hip/hip_bf16.h>

typedef __attribute__((ext_vector_type(2))) float v2f;
typedef __attribute__((ext_vector_type(8))) float v8f;

#define D 128

// ---------------------------------------------------------------------------
// Zero a region (float4 granularity)
// ---------------------------------------------------------------------------
__global__ __launch_bounds__(256) void zero_kernel(float4* __restrict__ p, long n4) {
    long t = (long)blockIdx.x * blockDim.x + threadIdx.x;
    if (t < n4) p[t] = make_float4(0.f, 0.f, 0.f, 0.f);
}

// ---------------------------------------------------------------------------
// Edge scatter: 32 lanes per edge, each lane moves float4 (512B coalesced row)
// agg[dst] += feat[src]; optionally cnt[dst] += 1
// ---------------------------------------------------------------------------
__global__ __launch_bounds__(256)
void scatter_kernel(const float* __restrict__ feat,
                    const long long* __restrict__ src,
                    const long long* __restrict__ dst,
                    float* __restrict__ agg, float* __restrict__ cnt,
                    int E, int doCount) {
    long t = (long)blockIdx.x * blockDim.x + threadIdx.x;
    int e = (int)(t >> 5);
    int lane = (int)(t & 31);
    if (e >= E) return;
    int s = (int)src[e];
    int d = (int)dst[e];
    float4 v = ((const float4*)(feat + (size_t)s * D))[lane];
    float* a = agg + (size_t)d * D + lane * 4;
    atomicAdd(a + 0, v.x);
    atomicAdd(a + 1, v.y);
    atomicAdd(a + 2, v.z);
    atomicAdd(a + 3, v.w);
    if (doCount && lane == 0) atomicAdd(&cnt[d], 1.0f);
}

// ---------------------------------------------------------------------------
// Fused SAGE layer (normalization + dual-GEMM + bias + optional ReLU):
//   out = (relu?)( (agg/max(cnt,1)) @ Wl + xin @ Wr + bias )
// One wave computes a 16-row x 128-col output tile using
// V_WMMA_F32_16X16X4_F32 (fp32 in, fp32 accumulate). The mean normalization
// is folded into the A-fragment load (lowers to one v_pk_mul_f32 per K-step,
// co-executing with the WMMAs).
//
// A fragment (16x4 f32): lane l holds A[M = l&15][K = kbase + (l>>4)*2 + {0,1}]
// B fragment (4x16 f32): lane l holds B[K = kbase + (l>>4)*2 + {0,1}][N = l&15]
// C/D (16x16 f32, 8 VGPRs): vgpr v, lane l -> M = v + (l>>4)*8, N = l&15
// ---------------------------------------------------------------------------
template <bool RELU>
__global__ __launch_bounds__(256)
void sage_gemm_kernel(const float* __restrict__ agg, const float* __restrict__ cnt,
                      const float* __restrict__ xin,
                      const float* __restrict__ Wl, const float* __restrict__ Wr,
                      const float* __restrict__ bias, float* __restrict__ out, int N) {
    const int lane = threadIdx.x & 31;
    const int wave = threadIdx.x >> 5;
    const int row0 = (blockIdx.x * 8 + wave) * 16;
    if (row0 >= N) return;  // wave-uniform: EXEC stays all-ones for live waves

    const int nm    = lane & 15;          // M index for A, N index for B/C/D
    const int khalf = (lane >> 4) << 1;   // 0 or 2: upper half-wave holds K+2
    const int mhi   = (lane >> 4) << 3;   // 0 or 8: C/D row offset

    int arow = row0 + nm;
    if (arow >= N) arow = N - 1;          // clamp (stores are guarded below)
    const float* __restrict__ aRow = agg + (size_t)arow * D;
    const float* __restrict__ xRow = xin + (size_t)arow * D;
    const float inv = 1.0f / fmaxf(cnt[arow], 1.0f);  // per-lane mean scale

    // Preload bias into accumulators (bias depends only on N column).
    v8f c[8];
#pragma unroll
    for (int j = 0; j < 8; ++j) {
        float bv = bias[j * 16 + nm];
#pragma unroll
        for (int v = 0; v < 8; ++v) c[j][v] = bv;
    }

    // K loop: 32 steps of K=4; A loaded once per step, reused across 8 N-tiles.
#pragma unroll 1
    for (int k = 0; k < D; k += 4) {
        const int kb = k + khalf;
        v2f a1, a2;
        a1.x = aRow[kb] * inv;            // fused mean normalization
        a1.y = aRow[kb + 1] * inv;
        a2.x = xRow[kb];
        a2.y = xRow[kb + 1];
#pragma unroll
        for (int j = 0; j < 8; ++j) {
            const int n = j * 16 + nm;
            v2f b1, b2;
            b1.x = Wl[(size_t)kb * D + n];
            b1.y = Wl[(size_t)(kb + 1) * D + n];
            b2.x = Wr[(size_t)kb * D + n];
            b2.y = Wr[(size_t)(kb + 1) * D + n];
            c[j] = __builtin_amdgcn_wmma_f32_16x16x4_f32(
                false, a1, false, b1, (short)0, c[j], false, false);
            c[j] = __builtin_amdgcn_wmma_f32_16x16x4_f32(
                false, a2, false, b2, (short)0, c[j], false, false);
        }
    }

    // Epilogue: optional ReLU + store.
    // Fast path (wave-uniform): all 16 rows in range -> straight-line stores,
    // EXEC stays all-ones, no per-element masking. Only the single partial
    // tile at the grid tail takes the guarded path.
    float* __restrict__ outBase = out + (size_t)(row0 + mhi) * D + nm;
    if (row0 + 16 <= N) {
#pragma unroll
        for (int j = 0; j < 8; ++j) {
#pragma unroll
            for (int v = 0; v < 8; ++v) {
                float val = c[j][v];
                if (RELU) val = fmaxf(val, 0.f);
                outBase[(size_t)v * D + j * 16] = val;
            }
        }
    } else {
#pragma unroll
        for (int j = 0; j < 8; ++j) {
#pragma unroll
            for (int v = 0; v < 8; ++v) {
                const int row = row0 + mhi + v;
                if (row < N) {
                    float val = c[j][v];
                    if (RELU) val = fmaxf(val, 0.f);
                    out[(size_t)row * D + j * 16 + nm] = val;
                }
            }
        }
    }
}

// ---------------------------------------------------------------------------
extern "C" void kernel_launch(void* const* d_in, const int* in_sizes, int n_in,
                              void* d_out, int out_size, void* d_ws, size_t ws_size,
                              hipStream_t stream) {
    const float* x        = (const float*)d_in[0];
    const long long* ei   = (const long long*)d_in[1];
    const float* Wl1      = (const float*)d_in[2];
    const float* Wr1      = (const float*)d_in[3];
    const float* b1       = (const float*)d_in[4];
    const float* Wl2      = (const float*)d_in[5];
    const float* Wr2      = (const float*)d_in[6];
    const float* b2       = (const float*)d_in[7];
    float* out            = (float*)d_out;

    const int N = in_sizes[0] / D;   // 50000
    const int E = in_sizes[1] / 2;   // 600000
    const long long* src = ei;       // edge_index[0]
    const long long* dst = ei + E;   // edge_index[1]

    // Workspace layout: cnt (padded) | agg | h
    const size_t padN = (size_t)((N + 255) & ~255);
    float* cnt = (float*)d_ws;
    float* agg = cnt + padN;
    float* h   = agg + (size_t)N * D;

    const int ZB = 256;
    const long n4_cnt_agg = (long)((padN + (size_t)N * D) / 4);
    const long n4_agg     = (long)(((size_t)N * D) / 4);
    const dim3 blk(256);
    const dim3 gridScatter((unsigned)(((long)E * 32 + 255) / 256));
    const dim3 gridGemm((unsigned)((N + 127) / 128));

    // ---------------- Layer 1 ----------------
    zero_kernel<<<(unsigned)((n4_cnt_agg + ZB - 1) / ZB), ZB, 0, stream>>>((float4*)cnt, n4_cnt_agg);
    scatter_kernel<<<gridScatter, blk, 0, stream>>>(x, src, dst, agg, cnt, E, 1);
    sage_gemm_kernel<true><<<gridGemm, blk, 0, stream>>>(agg, cnt, x, Wl1, Wr1, b1, h, N);

    // ---------------- Layer 2 ----------------
    zero_kernel<<<(unsigned)((n4_agg + ZB - 1) / ZB), ZB, 0, stream>>>((float4*)agg, n4_agg);
    scatter_kernel<<<gridScatter, blk, 0, stream>>>(h, src, dst, agg, cnt, E, 0);
    sage_gemm_kernel<false><<<gridGemm, blk, 0, stream>>>(agg, cnt, h, Wl2, Wr2, b2, out, N);
}